// MLE_16295105921491
// MI455X (gfx1250) — compile-verified
//
#include <hip/hip_runtime.h>
#include <hip/hip_bf16.h>

#define EPS_F      1e-7f
#define AGE_MAX_F  120.0f
// 0.5 * log(2*pi)
#define HALF_LOG_2PI 0.91893853320467274178f
#define SQRT2_F      1.41421356237309504880f

typedef __attribute__((ext_vector_type(2))) float v2f;
typedef __attribute__((ext_vector_type(8))) float v8f;

// ---------- per-element loss, interval branch (use_intvl != 0) ----------
__device__ __forceinline__ float loss_intvl(float mu, float s, float tte, float age) {
    float sigma     = __expf(s);
    float inv_denom = __frsqrt_rn(2.0f * sigma);          // 1/sqrt(2*sigma)
    float max_tte   = AGE_MAX_F - age;
    float cdf_max   = 0.5f + 0.5f * erff((__logf(max_tte) - mu) * inv_denom);
    float cdf_t     = 0.5f + 0.5f * erff((__logf(tte)     - mu) * inv_denom);
    return -__logf(cdf_max - cdf_t + EPS_F);
}

// ---------- per-element loss, MLE branch (use_intvl == 0) ----------
__device__ __forceinline__ float loss_mle(float mu, float s, float tte, float is_alive) {
    float sigma      = __expf(s);
    float inv_ssqrt2 = 1.0f / (sigma * SQRT2_F);
    float cdf_t      = 0.5f + 0.5f * erff((__logf(tte) - mu) * inv_ssqrt2);
    float alive_loss = -__logf(1.0f - cdf_t + EPS_F);
    float lt         = __logf(tte + EPS_F);
    float d          = lt - mu;
    float inv_s2     = __expf(-2.0f * s);                 // 1/sigma^2
    // log_prob = -lt - s - 0.5*log(2pi) - d^2/(2 sigma^2);  dead_loss = -log_prob
    float dead_loss  = lt + s + HALF_LOG_2PI + 0.5f * d * d * inv_s2;
    return is_alive * alive_loss + (1.0f - is_alive) * dead_loss;
}

// ---------- stage 1: streaming partial sums (grid-stride over element PAIRS) ----------
__global__ void __launch_bounds__(256)
mle_loss_partial(const float4* __restrict__ pred,   // (N,2) -> float4 = 2 elems
                 const float4* __restrict__ tgt,    // (N,2) -> float4 = 2 elems
                 const float2* __restrict__ ages,   // (N,)  -> float2 = 2 elems
                 const int*    __restrict__ use_intvl_p,
                 float*        __restrict__ block_sums,
                 int nPairs) {
    const int use_intvl = use_intvl_p[0];             // uniform device read (graph-safe)
    const int stride    = gridDim.x * blockDim.x;
    float acc = 0.0f;

    for (int p = blockIdx.x * blockDim.x + threadIdx.x; p < nPairs; p += stride) {
        // Hint the next grid-stride iteration into cache (emits global_prefetch_b8;
        // speculative prefetches past the end are silently dropped).
        __builtin_prefetch(&pred[p + stride], 0, 0);
        __builtin_prefetch(&tgt [p + stride], 0, 0);
        __builtin_prefetch(&ages[p + stride], 0, 0);

        float4 pp = pred[p];    // (mu0, s0, mu1, s1)        b128 load
        float4 tg = tgt [p];    // (tte0, alive0, tte1, alive1)
        float2 ag = ages[p];    // (age0, age1)

        if (use_intvl) {
            acc += loss_intvl(pp.x, pp.y, tg.x, ag.x);
            acc += loss_intvl(pp.z, pp.w, tg.z, ag.y);
        } else {
            acc += loss_mle(pp.x, pp.y, tg.x, tg.y);
            acc += loss_mle(pp.z, pp.w, tg.z, tg.w);
        }
    }

    // Block reduction: 256 threads (8 wave32) -> 1 partial, via LDS tree.
    __shared__ float sdata[256];
    sdata[threadIdx.x] = acc;
    __syncthreads();
#pragma unroll
    for (int off = 128; off > 0; off >>= 1) {
        if (threadIdx.x < off) sdata[threadIdx.x] += sdata[threadIdx.x + off];
        __syncthreads();
    }
    if (threadIdx.x == 0) block_sums[blockIdx.x] = sdata[0];
}

// ---------- stage 2: deterministic final fold; last 32->1 uses the matrix pipe ----------
__global__ void __launch_bounds__(256)
mle_loss_final(const float* __restrict__ block_sums,
               float*       __restrict__ out,
               int nb, float invN) {
    __shared__ float sdata[256];
    float s = 0.0f;
    for (int i = threadIdx.x; i < nb; i += 256) s += block_sums[i];
    sdata[threadIdx.x] = s;
    __syncthreads();

    if (threadIdx.x < 32) {                 // wave 0, EXEC all-1s inside this branch
        float v = 0.0f;
#pragma unroll
        for (int j = 0; j < 8; ++j) v += sdata[threadIdx.x + 32 * j];

        // WMMA reduction gadget: A is 16x4 f32.  Per the ISA layout, A VGPR0 holds
        // lanes 0-15 -> (M=lane, K=0) and lanes 16-31 -> (M=lane-16, K=2); A VGPR1
        // (K=1/K=3) is zeroed.  With B = ones(4x16):
        //   D[m][n] = A[m][0] + A[m][2] = v[m] + v[m+16]   (independent of n)
        v2f a; a[0] = v;    a[1] = 0.0f;
        v2f b; b[0] = 1.0f; b[1] = 1.0f;
        v8f c = {};
        c = __builtin_amdgcn_wmma_f32_16x16x4_f32(
                /*neg_a=*/false, a, /*neg_b=*/false, b,
                /*c_mod=*/(short)0, c, /*reuse_a=*/false, /*reuse_b=*/false);

        // D VGPR r at lane L: M = (L<16 ? r : r+8), N = L%16.
        // Lane 0 sums row-sums M=0..7; lane 16 sums row-sums M=8..15.
        float t = 0.0f;
#pragma unroll
        for (int j = 0; j < 8; ++j) t += c[j];
        float t16 = __shfl(t, 16, 32);      // wave32 shuffle
        if (threadIdx.x == 0) out[0] = (t + t16) * invN;
    }
}

extern "C" void kernel_launch(void* const* d_in, const int* in_sizes, int n_in,
                              void* d_out, int out_size, void* d_ws, size_t ws_size,
                              hipStream_t stream) {
    const float4* pred = (const float4*)d_in[0];   // (N,2) f32
    const float4* tgt  = (const float4*)d_in[1];   // (N,2) f32
    const float2* ages = (const float2*)d_in[2];   // (N,)  f32
    const int*    ui   = (const int*)  d_in[3];    // scalar int

    const int n      = in_sizes[2];                // N (= ages count)
    const int nPairs = n >> 1;                     // N is even (16777216)

    float* block_sums = (float*)d_ws;              // 2048 floats = 8 KB scratch

    const int NB = 2048;                           // 2048 partials
    const int NT = 256;                            // 8 wave32 per block

    mle_loss_partial<<<NB, NT, 0, stream>>>(pred, tgt, ages, ui, block_sums, nPairs);
    mle_loss_final  <<<1,  NT, 0, stream>>>(block_sums, (float*)d_out, NB,
                                            1.0f / (float)n);
}